// MyEmbedding_22978075033999
// MI455X (gfx1250) — compile-verified
//
#include <hip/hip_runtime.h>
#include <stdint.h>

// Embedding gather expressed the MI455X way: Tensor Data Mover gather-mode
// descriptors DMA 8 embedding rows (32 KB) per instruction HBM->LDS, then a
// flat 1-D descriptor streams the tile LDS->HBM. Pure data movement, ~64 MiB
// total -> ~2.8 us at 23.3 TB/s. No WMMA: the one-hot matmul form would be
// ~537 GFLOP of wasted math for a memory-bound gather.

#define VOCAB 32000
#define DIM   1024
#define BATCH 4
#define SEQ   2048
#define NTOK  (BATCH * SEQ)            /* 8192 tokens            */
#define ROWS_PER_DESC 8                /* 32-bit gather indices  */
#define NDESC (NTOK / ROWS_PER_DESC)   /* 1024 descriptors       */

typedef uint32_t v4u __attribute__((ext_vector_type(4)));
typedef int      v4i __attribute__((ext_vector_type(4)));
typedef int      v8i __attribute__((ext_vector_type(8)));

#if defined(__has_builtin)
#if __has_builtin(__builtin_amdgcn_tensor_load_to_lds) && \
    __has_builtin(__builtin_amdgcn_tensor_store_from_lds) && \
    __has_builtin(__builtin_amdgcn_s_wait_tensorcnt)
#define USE_TDM 1
#endif
#endif
#ifndef USE_TDM
#define USE_TDM 0
#endif

#if USE_TDM
// ROCm 7.2 (clang-22): 5-arg builtin. amdgpu-toolchain (clang-23): 6-arg.
#if defined(__clang_major__) && (__clang_major__ >= 23)
#define TDM_LOAD(a, b, c, d)  __builtin_amdgcn_tensor_load_to_lds((a), (b), (c), (d), (v8i){0,0,0,0,0,0,0,0}, 0)
#define TDM_STORE(a, b, c, d) __builtin_amdgcn_tensor_store_from_lds((a), (b), (c), (d), (v8i){0,0,0,0,0,0,0,0}, 0)
#else
#define TDM_LOAD(a, b, c, d)  __builtin_amdgcn_tensor_load_to_lds((a), (b), (c), (d), 0)
#define TDM_STORE(a, b, c, d) __builtin_amdgcn_tensor_store_from_lds((a), (b), (c), (d), 0)
#endif

__global__ __launch_bounds__(32) void embed_gather_tdm(
    const int* __restrict__ tok, const float* __restrict__ emb,
    float* __restrict__ out) {
  // 8 rows x 1024 f32 = 32 KB staging tile in LDS.
  __shared__ float lbuf[ROWS_PER_DESC * DIM];

  const uint32_t g  = blockIdx.x;            // descriptor id
  const uint32_t t0 = g * ROWS_PER_DESC;     // first token of this group

  // Flat LDS addresses carry the byte offset in the low 32 bits (aperture
  // distinguishing bits live in addr[63:32]).
  const uint32_t lds_addr = (uint32_t)(uintptr_t)(&lbuf[0]);

  // Uniform (blockIdx-derived) addresses -> scalar loads -> SGPR row indices.
  const int r0 = tok[t0 + 0];
  const int r1 = tok[t0 + 1];
  const int r2 = tok[t0 + 2];
  const int r3 = tok[t0 + 3];
  const int r4 = tok[t0 + 4];
  const int r5 = tok[t0 + 5];
  const int r6 = tok[t0 + 6];
  const int r7 = tok[t0 + 7];

  // ---- Gather-mode load descriptor: 8 rows of embedding -> LDS ----
  const uint64_t embA = (uint64_t)(uintptr_t)emb;
  v4u g0l;
  g0l.x = 0xC0000001u;                       // count=1 | idx32(b30) | gather(b31)
  g0l.y = lds_addr;                          // lds_addr[31:0]
  g0l.z = (uint32_t)embA;                    // global_addr[31:0]
  g0l.w = (uint32_t)(embA >> 32) | 0x80000000u; // global_addr[56:32] | type=2

  v8i g1l;
  g1l[0] = 0x00020000;                       // data_size = 2 (4 bytes)
  g1l[1] = (int)((uint32_t)DIM << 16);       // tensor_dim0 lo16 = 1024
  g1l[2] = (int)((uint32_t)VOCAB << 16);     // tensor_dim1 lo16 = 32000
  g1l[3] = (int)((uint32_t)DIM << 16);       // tile_dim0 = 1024 (row width)
  g1l[4] = ROWS_PER_DESC;                    // tile_dim1 = #valid indices = 8
  g1l[5] = DIM;                              // tensor_dim0_stride = 1024
  g1l[6] = 0;
  g1l[7] = 0;

  v4i g2l = {r0, r1, r2, r3};                // row_index_0..3 (32-bit mode)
  v4i g3l = {r4, r5, r6, r7};                // row_index_4..7

  TDM_LOAD(g0l, g1l, g2l, g3l);
  __builtin_amdgcn_s_wait_tensorcnt((short)0);   // tile resident in LDS

  // ---- Flat 1-D store descriptor: LDS tile -> contiguous output block ----
  const uint64_t outA = (uint64_t)(uintptr_t)(out + (size_t)t0 * DIM);
  const uint32_t TOTE = ROWS_PER_DESC * DIM;     // 8192 elements

  v4u g0s;
  g0s.x = 0x00000001u;                       // count=1, normal mode
  g0s.y = lds_addr;
  g0s.z = (uint32_t)outA;
  g0s.w = (uint32_t)(outA >> 32) | 0x80000000u;  // type=2

  v8i g1s;
  g1s[0] = 0x00020000;                       // data_size = 4 bytes
  g1s[1] = (int)(TOTE << 16);                // tensor_dim0 = 8192
  g1s[2] = (int)(1u << 16);                  // tensor_dim1 = 1
  g1s[3] = (int)(TOTE << 16);                // tile_dim0 = 8192
  g1s[4] = 0;                                // tile_dim1 = 0 (1-D tile)
  g1s[5] = (int)TOTE;                        // tensor_dim0_stride
  g1s[6] = 0;
  g1s[7] = 0;

  v4i z4 = {0, 0, 0, 0};
  TDM_STORE(g0s, g1s, z4, z4);
  // S_ENDPGM performs an implicit wait-idle, draining TENSORcnt before the
  // wave (and its LDS) is released.
}
#endif  // USE_TDM

// Fallback / reference-quality direct gather: one block per token row,
// 256 lanes x float4 = 4 KB row copy with B128 loads/stores.
__global__ __launch_bounds__(256) void embed_gather_copy(
    const int* __restrict__ tok, const float* __restrict__ emb,
    float* __restrict__ out) {
  const int row = blockIdx.x;
  const int t   = tok[row];
  const float4* __restrict__ src =
      (const float4*)(emb + (size_t)t * DIM);
  float4* __restrict__ dst = (float4*)(out + (size_t)row * DIM);
  dst[threadIdx.x] = src[threadIdx.x];
}

extern "C" void kernel_launch(void* const* d_in, const int* in_sizes, int n_in,
                              void* d_out, int out_size, void* d_ws,
                              size_t ws_size, hipStream_t stream) {
  const int*   tok = (const int*)d_in[0];    // [4, 2048] int32
  const float* emb = (const float*)d_in[1];  // [32000, 1024] f32
  float*       out = (float*)d_out;          // [4, 2048, 1024] f32
  (void)in_sizes; (void)n_in; (void)out_size; (void)d_ws; (void)ws_size;

#if USE_TDM
  embed_gather_tdm<<<NDESC, 32, 0, stream>>>(tok, emb, out);
#else
  embed_gather_copy<<<NTOK, 256, 0, stream>>>(tok, emb, out);
#endif
}